// QAvgPool2d_24034636988578
// MI455X (gfx1250) — compile-verified
//
#include <hip/hip_runtime.h>
#include <stdint.h>

// QAvgPool2d: (64,128,112,112) f32 -> (64,128,56,56) f32, 2x2/s2 mean with
// bf16 round-trip after each accumulation step. Bandwidth-bound (~514 MB
// @ 23.3 TB/s => ~22 us floor). Data path: async global->LDS B128 staging
// (ASYNCcnt double buffer, NT hint) + conflict-free ds_load_b128 + NT stores.
// Address math strength-reduced (iterate along bc => constant strides);
// quantization via packed v_cvt_pk_bf16_f32 (RNE) to cut VALU pressure.

typedef float v2f __attribute__((ext_vector_type(2)));

#define BLOCK 256
#define NITER 8

// Geometry constants
#define IN_STEP_FLOATS  12845056u   // 1024 * C-plane (12544 floats) per k-step
#define OUT_STEP_PAIRS   1605632u   // 1024 * 1568 output pairs per k-step

// Dual bf16 quantize: (qa,qb) = f32(bf16(a)), f32(bf16(b)) with RNE.
// v_cvt_pk_bf16_f32: D[15:0]=bf16(S0), D[31:16]=bf16(S1).
__device__ __forceinline__ void qpk(float a, float b, float& qa, float& qb) {
  uint32_t d;
  asm("v_cvt_pk_bf16_f32 %0, %1, %2" : "=v"(d) : "v"(a), "v"(b));
  qa = __uint_as_float(d << 16);
  qb = __uint_as_float(d & 0xFFFF0000u);
}

__global__ __launch_bounds__(BLOCK, 1) void qavgpool2d_async(
    const float* __restrict__ x, float* __restrict__ y) {
  // Per-thread 16B slots, stride 16 -> lanes 0..15 cover all 64 LDS banks
  // on ds_load_b128 readback (conflict-free). 16 KB total.
  __shared__ float4 sTop[2][BLOCK];
  __shared__ float4 sBot[2][BLOCK];

  const uint32_t tid = threadIdx.x;
  uint32_t t = blockIdx.x * BLOCK + tid;

  // One-time decomposition: t -> (ow, oh, bc_lo); iterations walk bc.
  uint32_t ow = t % 28u;        // float4 index within input row (pair index within output row)
  uint32_t r  = t / 28u;
  uint32_t oh = r % 56u;
  uint32_t bc = r / 56u;        // 0..1023; full bc = bc + 1024*k

  const float* g0 = x + ((size_t)bc * 12544u + (size_t)oh * 224u + (size_t)ow * 4u);
  v2f*         o0 = (v2f*)y + ((size_t)bc * 1568u + (size_t)oh * 28u + (size_t)ow);

  // LDS byte offsets (generic pointer low 32 bits == LDS offset)
  const uint32_t lt0 = (uint32_t)(uintptr_t)&sTop[0][tid];
  const uint32_t lb0 = (uint32_t)(uintptr_t)&sBot[0][tid];
  const uint32_t lt1 = (uint32_t)(uintptr_t)&sTop[1][tid];
  const uint32_t lb1 = (uint32_t)(uintptr_t)&sBot[1][tid];

  // Stage one output-pair's inputs: float4 from row 2*oh and row 2*oh+1
  // (rows are 448 B apart -> folded into IOFFSET). Input is streamed once
  // => non-temporal hint.
  auto stage = [&](const float* g, uint32_t lt, uint32_t lb) {
    uint64_t ga = (uint64_t)(uintptr_t)g;
    asm volatile("global_load_async_to_lds_b128 %0, %1, off th:TH_LOAD_NT"
                 :: "v"(lt), "v"(ga) : "memory");
    asm volatile("global_load_async_to_lds_b128 %0, %1, off offset:448 th:TH_LOAD_NT"
                 :: "v"(lb), "v"(ga) : "memory");
  };

  stage(g0, lt0, lb0);  // prime the pipeline

  const float* g = g0;
  v2f* op = o0;

#pragma unroll
  for (int it = 0; it < NITER; ++it) {
    // Prefetch next bc-tile into the other buffer; final iteration re-fetches
    // the base tile (valid address, keeps loop branch-free, EXEC all-ones).
    const float* gn = (it + 1 < NITER) ? (g + (size_t)IN_STEP_FLOATS) : g0;
    if ((it & 1) == 0) stage(gn, lt1, lb1);
    else               stage(gn, lt0, lb0);

    // Async loads retire in order: <=2 outstanding => previous stage landed.
    asm volatile("s_wait_asynccnt 0x2" ::: "memory");

    float4 tv = sTop[it & 1][tid];   // ds_load_b128
    float4 bv = sBot[it & 1][tid];   // ds_load_b128

    // Reference order per output: q(x00); q(+x01); q(+x10); q(+x11); q(*0.25).
    // The two outputs' chains are independent -> packed dual quant.
    float s0, s1;
    qpk(tv.x,        tv.z,        s0, s1);
    qpk(s0 + tv.y,   s1 + tv.w,   s0, s1);
    qpk(s0 + bv.x,   s1 + bv.z,   s0, s1);
    qpk(s0 + bv.y,   s1 + bv.w,   s0, s1);
    qpk(s0 * 0.25f,  s1 * 0.25f,  s0, s1);

    v2f o; o.x = s0; o.y = s1;
    __builtin_nontemporal_store(o, op);  // write-once output

    g  += (size_t)IN_STEP_FLOATS;
    op += (size_t)OUT_STEP_PAIRS;
  }
}

extern "C" void kernel_launch(void* const* d_in, const int* in_sizes, int n_in,
                              void* d_out, int out_size, void* d_ws, size_t ws_size,
                              hipStream_t stream) {
  (void)in_sizes; (void)n_in; (void)d_ws; (void)ws_size;
  const float* x = (const float*)d_in[0];
  float* y = (float*)d_out;

  // out_size = 25,690,112 floats -> 12,845,056 pairs -> /NITER threads.
  const uint32_t npairs  = (uint32_t)(out_size / 2);
  const uint32_t threads = npairs / NITER;        // 1,605,632
  const uint32_t blocks  = threads / BLOCK;       // 6272, exact
  qavgpool2d_async<<<blocks, BLOCK, 0, stream>>>(x, y);
}